// MultiGATLayer_74277164417104
// MI455X (gfx1250) — compile-verified
//
#include <hip/hip_runtime.h>
#include <hip/hip_bf16.h>
#include <math.h>

// ---------------------------------------------------------------------------
// Multi-head GAT layer for MI455X (gfx1250, wave32).
//   N=50000 nodes, F_IN=128, F_OUT=32, H=4 heads, E=1.6M edges.
// GEMM uses native fp32 WMMA (v_wmma_f32_16x16x4_f32) -> full precision.
// Edge phases are L2-atomic bound; logits are recomputed per pass instead of
// materializing per-edge buffers (key/out stay L2-resident: 192MB L2).
// ---------------------------------------------------------------------------

typedef __attribute__((ext_vector_type(2))) float v2f;
typedef __attribute__((ext_vector_type(8))) float v8f;

#define FIN   128
#define FOUTH 128   // H * F_OUT
#define KC    32    // K chunk staged in LDS per iteration

// ---------------------------------------------------------------------------
// init: zero out & denom, set m = -inf
// ---------------------------------------------------------------------------
__global__ __launch_bounds__(256) void gat_init(float* __restrict__ out,
                                                float* __restrict__ m,
                                                float* __restrict__ denom,
                                                int nN) {
  int i = blockIdx.x * blockDim.x + threadIdx.x;
  if (i < nN * FOUTH) out[i] = 0.0f;
  if (i < nN * 4) {
    m[i]     = -INFINITY;
    denom[i] = 0.0f;
  }
}

// ---------------------------------------------------------------------------
// key = features @ W_w + W_b   (N x 128) = (N x 128)(128 x 128)
// 256 threads = 8 waves; each block computes a 128-row strip; each wave a
// 16-row strip across all 8 column tiles (8 x v8f accumulators).
// WMMA f32 16x16x4 fragment layouts (ISA 7.12.2, 32-bit operands):
//   A (16x4): lane<16 -> M=lane, regs {K0,K1}; lane>=16 -> M=lane-16, {K2,K3}
//   B (4x16): lane<16 -> N=lane, regs {K0,K1}; lane>=16 -> N=lane-16, {K2,K3}
//   C/D     : reg r, lane<16 -> M=r,N=lane ; lane>=16 -> M=8+r,N=lane-16
// Staging loads are batched into registers before the LDS stores (4 loads ->
// 1 wait -> 4 stores), and all 8 B fragments of a k-step are fetched before
// the 8 WMMAs so the ds-load latency is paid once per k-step, not per tile.
// ---------------------------------------------------------------------------
__global__ __launch_bounds__(256) void gat_gemm_key(const float* __restrict__ feat,
                                                    const float* __restrict__ Ww,
                                                    const float* __restrict__ Wb,
                                                    float* __restrict__ key,
                                                    int nN) {
  __shared__ __attribute__((aligned(16))) float Alds[128][KC + 2];   // [m][k]
  __shared__ __attribute__((aligned(16))) float Blds[KC / 2][2 * FOUTH + 2]; // [k/2][n*2+k&1]

  const int tid  = threadIdx.x;
  const int wave = tid >> 5;
  const int lane = tid & 31;
  const int lm   = lane & 15;          // M (for A) / N (for B,D) within tile
  const int lk   = (lane >> 4) << 1;   // K sub-offset: 0 or 2
  const int block_row = blockIdx.x * 128;

  const v8f vzero = {0.f, 0.f, 0.f, 0.f, 0.f, 0.f, 0.f, 0.f};
  v8f acc[8];
#pragma unroll
  for (int t = 0; t < 8; ++t) acc[t] = vzero;

  for (int kc = 0; kc < FIN; kc += KC) {
    // ---- stage A chunk: 128 rows x 32 cols. Batch 4 loads, then 4 stores.
    float4 va[4];
#pragma unroll
    for (int it = 0; it < 4; ++it) {
      int idx = tid + it * 256;
      int r   = idx >> 3;            // 8 float4 per row
      int c   = (idx & 7) << 2;
      int gr  = block_row + r;
      va[it] = make_float4(0.f, 0.f, 0.f, 0.f);
      if (gr < nN) va[it] = *(const float4*)(feat + (size_t)gr * FIN + kc + c);
    }
    // ---- stage B chunk: rows k=kc..kc+31 of Ww. Batch loads likewise.
    float4 vb[4];
#pragma unroll
    for (int it = 0; it < 4; ++it) {
      int idx = tid + it * 256;
      int kr  = idx >> 5;            // 32 float4 per k-row
      int c   = (idx & 31) << 2;
      vb[it] = *(const float4*)(Ww + (size_t)(kc + kr) * FOUTH + c);
    }
#pragma unroll
    for (int it = 0; it < 4; ++it) {
      int idx = tid + it * 256;
      int r   = idx >> 3;
      int c   = (idx & 7) << 2;
      Alds[r][c + 0] = va[it].x; Alds[r][c + 1] = va[it].y;
      Alds[r][c + 2] = va[it].z; Alds[r][c + 3] = va[it].w;
    }
#pragma unroll
    for (int it = 0; it < 4; ++it) {
      int idx = tid + it * 256;
      int kr  = idx >> 5;
      int c   = (idx & 31) << 2;
      int kk = kr >> 1, par = kr & 1;
      Blds[kk][(c + 0) * 2 + par] = vb[it].x;
      Blds[kk][(c + 1) * 2 + par] = vb[it].y;
      Blds[kk][(c + 2) * 2 + par] = vb[it].z;
      Blds[kk][(c + 3) * 2 + par] = vb[it].w;
    }
    __syncthreads();

#pragma unroll
    for (int ks = 0; ks < KC; ks += 4) {
      v2f a = *(const v2f*)&Alds[wave * 16 + lm][ks + lk];
      const int kk = (ks + lk) >> 1;
      v2f b[8];
#pragma unroll
      for (int t = 0; t < 8; ++t)
        b[t] = *(const v2f*)&Blds[kk][(t * 16 + lm) * 2];
#pragma unroll
      for (int t = 0; t < 8; ++t) {
        acc[t] = __builtin_amdgcn_wmma_f32_16x16x4_f32(
            /*neg_a=*/false, a, /*neg_b=*/false, b[t],
            /*c_mod=*/(short)0, acc[t], /*reuse_a=*/false, /*reuse_b=*/false);
      }
    }
    __syncthreads();
  }

  // ---- epilogue: D[r,lane] -> key, fuse bias ----
  const int mrow0 = block_row + wave * 16 + ((lane >> 4) << 3);
#pragma unroll
  for (int t = 0; t < 8; ++t) {
    int col    = t * 16 + lm;
    float bias = Wb[col];
#pragma unroll
    for (int r = 0; r < 8; ++r) {
      int mr = mrow0 + r;
      if (mr < nN) key[(size_t)mr * FOUTH + col] = acc[t][r] + bias;
    }
  }
}

// ---------------------------------------------------------------------------
// s_dst[n,h] = key[n,h,:] . a_w[h,:32]   s_src[n,h] = key[n,h,:] . a_w[h,32:]
// ---------------------------------------------------------------------------
__global__ __launch_bounds__(256) void gat_scores(const float* __restrict__ key,
                                                  const float* __restrict__ aw,
                                                  float* __restrict__ s_dst,
                                                  float* __restrict__ s_src,
                                                  int nN) {
  int i = blockIdx.x * blockDim.x + threadIdx.x;
  int node = i >> 2, h = i & 3;
  if (node >= nN) return;
  const float* kp = key + (size_t)node * FOUTH + h * 32;
  const float* ah = aw + h * 64;
  float sd = 0.f, ss = 0.f;
#pragma unroll
  for (int f = 0; f < 32; ++f) {
    float kv = kp[f];
    sd += kv * ah[f];
    ss += kv * ah[32 + f];
  }
  s_dst[node * 4 + h] = sd;
  s_src[node * 4 + h] = ss;
}

// float atomic max via signed/unsigned int monotonicity trick
__device__ __forceinline__ void atomicMaxF(float* addr, float v) {
  if (v >= 0.f) atomicMax((int*)addr, __float_as_int(v));
  else          atomicMin((unsigned int*)addr, __float_as_uint(v));
}

__device__ __forceinline__ float lrelu02(float x) {
  return x >= 0.f ? x : 0.2f * x;
}

// ---------------------------------------------------------------------------
// Pass 1: segment max of leaky-relu logits. One thread per edge slot
// (slots >= E are self loops). 4 heads per thread.
// ---------------------------------------------------------------------------
__global__ __launch_bounds__(256) void gat_edge_max(const int* __restrict__ esrc,
                                                    const int* __restrict__ edst,
                                                    const float* __restrict__ s_dst,
                                                    const float* __restrict__ s_src,
                                                    const float* __restrict__ ab,
                                                    float* __restrict__ m,
                                                    int nE, int nN) {
  int i = blockIdx.x * blockDim.x + threadIdx.x;
  int total = nE + nN;
  if (i >= total) return;
  int s = (i < nE) ? esrc[i] : (i - nE);
  int d = (i < nE) ? edst[i] : (i - nE);
#pragma unroll
  for (int h = 0; h < 4; ++h) {
    float e = lrelu02(s_dst[d * 4 + h] + s_src[s * 4 + h] + ab[h]);
    atomicMaxF(&m[d * 4 + h], e);
  }
}

// ---------------------------------------------------------------------------
// Pass 2: denom = segment_sum(exp(e - m[dst])). Logits recomputed (cheaper
// than round-tripping a 26MB per-edge buffer through L2).
// ---------------------------------------------------------------------------
__global__ __launch_bounds__(256) void gat_edge_denom(const int* __restrict__ esrc,
                                                      const int* __restrict__ edst,
                                                      const float* __restrict__ s_dst,
                                                      const float* __restrict__ s_src,
                                                      const float* __restrict__ ab,
                                                      const float* __restrict__ m,
                                                      float* __restrict__ denom,
                                                      int nE, int nN) {
  int i = blockIdx.x * blockDim.x + threadIdx.x;
  int total = nE + nN;
  if (i >= total) return;
  int s = (i < nE) ? esrc[i] : (i - nE);
  int d = (i < nE) ? edst[i] : (i - nE);
#pragma unroll
  for (int h = 0; h < 4; ++h) {
    float e = lrelu02(s_dst[d * 4 + h] + s_src[s * 4 + h] + ab[h]);
    atomicAdd(&denom[d * 4 + h], __expf(e - m[d * 4 + h]));
  }
}

// ---------------------------------------------------------------------------
// Pass 3: out[dst] += alpha * key[src]. One wave per edge: lane l owns
// floats [4l, 4l+4) of the 128-wide row (head = l/8) -> the key[src] gather
// is one contiguous 512B float4 burst per wave; 4 fp32 atomics per lane.
// ---------------------------------------------------------------------------
__global__ __launch_bounds__(256) void gat_edge_aggr(const int* __restrict__ esrc,
                                                     const int* __restrict__ edst,
                                                     const float* __restrict__ s_dst,
                                                     const float* __restrict__ s_src,
                                                     const float* __restrict__ ab,
                                                     const float* __restrict__ m,
                                                     const float* __restrict__ denom,
                                                     const float* __restrict__ key,
                                                     float* __restrict__ out,
                                                     int nE, int nN) {
  int lane = threadIdx.x & 31;
  long long eidx = (long long)blockIdx.x * (blockDim.x >> 5) + (threadIdx.x >> 5);
  long long total = (long long)nE + nN;
  if (eidx >= total) return;
  int s = (eidx < nE) ? esrc[eidx] : (int)(eidx - nE);
  int d = (eidx < nE) ? edst[eidx] : (int)(eidx - nE);

  int h = lane >> 3;  // 8 lanes (32 floats) per head
  float e = lrelu02(s_dst[d * 4 + h] + s_src[s * 4 + h] + ab[h]);
  float alpha = __expf(e - m[d * 4 + h]) / denom[d * 4 + h];

  float4 kv = ((const float4*)(key + (size_t)s * FOUTH))[lane];
  float* op = out + (size_t)d * FOUTH + lane * 4;
  atomicAdd(op + 0, alpha * kv.x);
  atomicAdd(op + 1, alpha * kv.y);
  atomicAdd(op + 2, alpha * kv.z);
  atomicAdd(op + 3, alpha * kv.w);
}

// ---------------------------------------------------------------------------
extern "C" void kernel_launch(void* const* d_in, const int* in_sizes, int n_in,
                              void* d_out, int out_size, void* d_ws, size_t ws_size,
                              hipStream_t stream) {
  const float* feat = (const float*)d_in[0];
  const int*   esrc = (const int*)  d_in[1];
  const int*   edst = (const int*)  d_in[2];
  const float* Ww   = (const float*)d_in[3];
  const float* Wb   = (const float*)d_in[4];
  const float* aw   = (const float*)d_in[5];
  const float* ab   = (const float*)d_in[6];

  const int nN = in_sizes[0] / FIN;   // 50000
  const int nE = in_sizes[1];         // 1600000

  float* out   = (float*)d_out;                    // N x 128
  float* ws    = (float*)d_ws;
  float* key   = ws;                               // N x 128
  float* s_dst = key   + (size_t)nN * FOUTH;       // N x 4
  float* s_src = s_dst + (size_t)nN * 4;           // N x 4
  float* mbuf  = s_src + (size_t)nN * 4;           // N x 4
  float* denom = mbuf  + (size_t)nN * 4;           // N x 4

  const int total = nE + nN;

  gat_init<<<(nN * FOUTH + 255) / 256, 256, 0, stream>>>(out, mbuf, denom, nN);

  gat_gemm_key<<<(nN + 127) / 128, 256, 0, stream>>>(feat, Ww, Wb, key, nN);

  gat_scores<<<(nN * 4 + 255) / 256, 256, 0, stream>>>(key, aw, s_dst, s_src, nN);

  gat_edge_max<<<(total + 255) / 256, 256, 0, stream>>>(
      esrc, edst, s_dst, s_src, ab, mbuf, nE, nN);

  gat_edge_denom<<<(total + 255) / 256, 256, 0, stream>>>(
      esrc, edst, s_dst, s_src, ab, mbuf, denom, nE, nN);

  gat_edge_aggr<<<(total + 7) / 8, 256, 0, stream>>>(
      esrc, edst, s_dst, s_src, ab, mbuf, denom, key, out, nE, nN);
}